// RaftModule_25692494365423
// MI455X (gfx1250) — compile-verified
//
#include <hip/hip_runtime.h>
#include <hip/hip_bf16.h>

// Problem constants (B=1, C=256, H=64, W=128, radius=4)
#define CDIM 256
#define HH   64
#define WW   128
#define HW   8192         // H*W
#define KT   32           // K tile staged in LDS per iteration
#define BM   64           // block tile M (queries)
#define BN   128          // block tile N (targets)

typedef float v2f __attribute__((ext_vector_type(2)));
typedef float v8f __attribute__((ext_vector_type(8)));

// ---------------------------------------------------------------------------
// Pass 1: corr_slab[q_local, t] = (1/16) * sum_c fmap1[c, q_base+q_local] * fmap2[c, t]
// fp32 WMMA 16x16x4, 8 waves/block, each wave owns a 16x64 strip (4 acc tiles).
// ---------------------------------------------------------------------------
__global__ __launch_bounds__(256) void corr_gemm_wmma(
    const float* __restrict__ A,      // fmap1 [C][HW]
    const float* __restrict__ B,      // fmap2 [C][HW]
    float* __restrict__ Cslab,        // [M_chunk][HW]
    int q_base)
{
    __shared__ float lds_a[BM * KT];  // [m][k]
    __shared__ float lds_b[BN * KT];  // [n][k]

    const int tid   = threadIdx.x;
    const int lane  = tid & 31;
    const int wave  = tid >> 5;       // 0..7
    const int wm    = wave >> 1;      // 0..3  (16-row M sub-tile)
    const int wn    = wave & 1;       // 0..1  (64-col N sub-tile)
    const int lhalf = lane >> 4;      // 0/1 : K pair select per WMMA layout
    const int lmod  = lane & 15;

    const int m0 = blockIdx.x * BM;   // query offset within chunk
    const int n0 = blockIdx.y * BN;   // target offset

    v8f acc[4];
    #pragma unroll
    for (int i = 0; i < 4; ++i)
        #pragma unroll
        for (int j = 0; j < 8; ++j)
            acc[i][j] = 0.0f;

    for (int k0 = 0; k0 < CDIM; k0 += KT) {
        __syncthreads();
        // Stage A tile: global rows are contiguous in q -> float4 loads,
        // store transposed [q][k] so fragments are contiguous k-pairs.
        #pragma unroll
        for (int it = 0; it < 2; ++it) {
            int f  = tid + it * 256;              // 512 float4s total
            int kr = f >> 4;                      // 0..31
            int qq = (f & 15) << 2;               // 0..60
            float4 v = *reinterpret_cast<const float4*>(
                A + (size_t)(k0 + kr) * HW + (q_base + m0 + qq));
            lds_a[(qq + 0) * KT + kr] = v.x;
            lds_a[(qq + 1) * KT + kr] = v.y;
            lds_a[(qq + 2) * KT + kr] = v.z;
            lds_a[(qq + 3) * KT + kr] = v.w;
        }
        // Stage B tile: 128 targets x 32 k
        #pragma unroll
        for (int it = 0; it < 4; ++it) {
            int f  = tid + it * 256;              // 1024 float4s total
            int kr = f >> 5;                      // 0..31
            int tt = (f & 31) << 2;               // 0..124
            float4 v = *reinterpret_cast<const float4*>(
                B + (size_t)(k0 + kr) * HW + (n0 + tt));
            lds_b[(tt + 0) * KT + kr] = v.x;
            lds_b[(tt + 1) * KT + kr] = v.y;
            lds_b[(tt + 2) * KT + kr] = v.z;
            lds_b[(tt + 3) * KT + kr] = v.w;
        }
        __syncthreads();

        const float* pa = &lds_a[(wm * 16 + lmod) * KT];
        #pragma unroll
        for (int kk = 0; kk < KT; kk += 4) {
            // A 16x4 fp32 fragment: lanes 0-15 hold (K,K+1), lanes 16-31 (K+2,K+3)
            v2f afrag = *reinterpret_cast<const v2f*>(pa + kk + 2 * lhalf);
            #pragma unroll
            for (int ns = 0; ns < 4; ++ns) {
                const float* pb = &lds_b[(wn * 64 + ns * 16 + lmod) * KT];
                v2f bfrag = *reinterpret_cast<const v2f*>(pb + kk + 2 * lhalf);
                acc[ns] = __builtin_amdgcn_wmma_f32_16x16x4_f32(
                    false, afrag, false, bfrag, (short)0, acc[ns], false, false);
            }
        }
    }

    // C/D layout: VGPR j -> row = base + j + 8*lhalf, col = base + (lane&15)
    const int row_base = m0 + wm * 16 + 8 * lhalf;
    #pragma unroll
    for (int ns = 0; ns < 4; ++ns) {
        const int col = n0 + wn * 64 + ns * 16 + lmod;
        #pragma unroll
        for (int j = 0; j < 8; ++j) {
            Cslab[(size_t)(row_base + j) * HW + col] = acc[ns][j] * 0.0625f;
        }
    }
}

// ---------------------------------------------------------------------------
// Pass 2: 9x9 bilinear lookup (zero padding) for the chunk's queries.
// Slab is L2-resident (32 MB << 192 MB) when this runs right after the GEMM.
// ---------------------------------------------------------------------------
__global__ __launch_bounds__(256) void corr_lookup_kernel(
    const float* __restrict__ coords,   // [2][HW] : x then y
    const float* __restrict__ slab,     // [nq][HW]
    float* __restrict__ out,            // [81][HW]
    int q_base, int nq)
{
    int t = blockIdx.x * blockDim.x + threadIdx.x;
    if (t >= nq * 81) return;
    int ql = t / 81;
    int kk = t - ql * 81;
    int q  = q_base + ql;

    float cx = coords[q];        // x
    float cy = coords[HW + q];   // y
    int di = kk / 9;             // dx index (meshgrid 'ij': first axis = dx)
    int dj = kk - di * 9;        // dy index
    float px = cx + (float)(di - 4);
    float py = cy + (float)(dj - 4);

    float x0f = floorf(px), y0f = floorf(py);
    float fx = px - x0f, fy = py - y0f;
    int x0 = (int)x0f, y0 = (int)y0f;

    const float* row = slab + (size_t)ql * HW;
    float g00 = 0.0f, g01 = 0.0f, g10 = 0.0f, g11 = 0.0f;
    bool yi0 = (y0 >= 0) & (y0 < HH);
    bool yi1 = (y0 + 1 >= 0) & (y0 + 1 < HH);
    bool xi0 = (x0 >= 0) & (x0 < WW);
    bool xi1 = (x0 + 1 >= 0) & (x0 + 1 < WW);
    if (yi0 && xi0) g00 = row[y0 * WW + x0];
    if (yi0 && xi1) g01 = row[y0 * WW + x0 + 1];
    if (yi1 && xi0) g10 = row[(y0 + 1) * WW + x0];
    if (yi1 && xi1) g11 = row[(y0 + 1) * WW + x0 + 1];

    float v = (1.0f - fy) * (1.0f - fx) * g00
            + (1.0f - fy) * fx          * g01
            + fy          * (1.0f - fx) * g10
            + fy          * fx          * g11;

    out[(size_t)kk * HW + q] = v;
}

// ---------------------------------------------------------------------------
extern "C" void kernel_launch(void* const* d_in, const int* in_sizes, int n_in,
                              void* d_out, int out_size, void* d_ws, size_t ws_size,
                              hipStream_t stream)
{
    const float* fmap1  = (const float*)d_in[0];
    const float* fmap2  = (const float*)d_in[1];
    const float* coords = (const float*)d_in[2];
    float* out  = (float*)d_out;
    float* slab = (float*)d_ws;

    // Pick the largest query-row chunk whose corr slab fits the workspace;
    // 8 rows -> 32 MB slab, which also stays resident in the 192 MB L2.
    int chunk_rows = 8;
    while ((size_t)chunk_rows * WW * HW * sizeof(float) > ws_size && chunk_rows > 1)
        chunk_rows >>= 1;
    const int mchunk = chunk_rows * WW;   // queries per chunk (multiple of BM)

    for (int q_base = 0; q_base < HW; q_base += mchunk) {
        dim3 g1(mchunk / BM, HW / BN);
        corr_gemm_wmma<<<g1, 256, 0, stream>>>(fmap1, fmap2, slab, q_base);

        int nt = mchunk * 81;
        corr_lookup_kernel<<<(nt + 255) / 256, 256, 0, stream>>>(
            coords, slab, out, q_base, mchunk);
    }
}